// edGNNLayer_36189394436357
// MI455X (gfx1250) — compile-verified
//
#include <hip/hip_runtime.h>
#include <hip/hip_bf16.h>

#define N_NODES 50000
#define N_EDGES 800000
// encoded -inf under the order-preserving float->uint map
#define NEG_ENC 0x007FFFFFu

typedef __attribute__((ext_vector_type(2))) float v2f;
typedef __attribute__((ext_vector_type(8))) float v8f;
typedef unsigned int u32x4 __attribute__((ext_vector_type(4)));
typedef int i32x8 __attribute__((ext_vector_type(8)));
typedef int i32x4 __attribute__((ext_vector_type(4)));

#if __has_builtin(__builtin_amdgcn_tensor_load_to_lds) && __has_builtin(__builtin_amdgcn_s_wait_tensorcnt)
#define USE_TDM 1
#else
#define USE_TDM 0
#endif

// order-preserving float <-> uint encoding (for atomicMax-based segment max)
__device__ __forceinline__ unsigned ford(float f) {
    unsigned u = __float_as_uint(f);
    return (u & 0x80000000u) ? ~u : (u | 0x80000000u);
}
__device__ __forceinline__ float funord(unsigned u) {
    unsigned v = (u & 0x80000000u) ? (u & 0x7FFFFFFFu) : ~u;
    return __uint_as_float(v);
}

__device__ __forceinline__ float wave_sum(float v) {
    #pragma unroll
    for (int off = 16; off > 0; off >>= 1) v += __shfl_xor(v, off, 32);
    return v;
}

// ---------------- K0: per-node init: softmax state + h_cat = [h | 0] ----------------
__global__ void k_init(const float* __restrict__ h, unsigned* __restrict__ m_src,
                       float* __restrict__ sum_src, unsigned* __restrict__ m_dst,
                       float* __restrict__ sum_dst, float* __restrict__ hcat, int N) {
    int n = blockIdx.x * blockDim.x + threadIdx.x;
    if (n >= N) return;
    m_src[n] = NEG_ENC; sum_src[n] = 0.f;
    m_dst[n] = NEG_ENC; sum_dst[n] = 0.f;
    const float4* hv = (const float4*)(h + (size_t)n * 64);
    float4* rv = (float4*)(hcat + (size_t)n * 160);
    #pragma unroll
    for (int i = 0; i < 16; ++i) rv[i] = hv[i];          // h_cat[:,0:64] = h
    float4 z = make_float4(0.f, 0.f, 0.f, 0.f);
    #pragma unroll
    for (int i = 16; i < 40; ++i) rv[i] = z;             // h_cat[:,64:160] = 0
}

// ---------------- K1: wave/edge group score a = z2 . w_group, segmax over src ----------------
__global__ void k_src_score(const float* __restrict__ h, const float* __restrict__ ef,
                            const int* __restrict__ src, const int* __restrict__ dst,
                            const float* __restrict__ wg, float* __restrict__ a_e,
                            unsigned* __restrict__ m_src, int E) {
    int e = blockIdx.x * (blockDim.x >> 5) + (threadIdx.x >> 5);
    int l = threadIdx.x & 31;
    if (e >= E) return;
    int s = src[e], d = dst[e];
    const float* hs = h + (size_t)s * 64;
    const float* hd = h + (size_t)d * 64;
    float acc = ef[(size_t)e * 32 + l] * wg[l];          // edge_dim = 32
    acc += hs[l]      * wg[32 + l];
    acc += hs[32 + l] * wg[64 + l];
    acc += hd[l]      * wg[96 + l];
    acc += hd[32 + l] * wg[128 + l];
    acc = wave_sum(acc);
    if (l == 0) {
        a_e[e] = acc;
        atomicMax(m_src + s, ford(acc));
    }
}

// ---------------- K2/K4: thread/edge: exp(score - segmax), segsum (shared for src & dst) ----------------
__global__ void k_seg_exp(const float* __restrict__ score, const int* __restrict__ seg,
                          const unsigned* __restrict__ m, float* __restrict__ exv,
                          float* __restrict__ sum, int E) {
    int e = blockIdx.x * blockDim.x + threadIdx.x;
    if (e >= E) return;
    int g = seg[e];
    float ex = __expf(score[e] - funord(m[g]));
    exv[e] = ex;
    atomicAdd(sum + g, ex);
}

// ---------------- K3: wave/edge: gamma, msg score s = msg . w_srcattn, segmax over dst ----------------
__global__ void k_msg_score(const float* __restrict__ h, const float* __restrict__ ef,
                            const int* __restrict__ src, const int* __restrict__ dst,
                            const float* __restrict__ wa, const float* __restrict__ sum_src,
                            float* __restrict__ g_e, float* __restrict__ s_e,
                            unsigned* __restrict__ m_dst, int E) {
    int e = blockIdx.x * (blockDim.x >> 5) + (threadIdx.x >> 5);
    int l = threadIdx.x & 31;
    if (e >= E) return;
    int s = src[e], d = dst[e];
    float gamma = g_e[e] / sum_src[s];
    const float* hs = h + (size_t)s * 64;
    float acc = hs[l] * wa[l] + hs[32 + l] * wa[32 + l]
              + gamma * ef[(size_t)e * 32 + l] * wa[64 + l];
    acc = wave_sum(acc);
    if (l == 0) {
        g_e[e] = gamma;              // overwrite exp with normalized gamma (read-before-write in-wave)
        s_e[e] = acc;
        atomicMax(m_dst + d, ford(acc));
    }
}

// ---------------- K5: wave/edge: accum[dst] += alpha * msg (96 dims, 3 atomics/lane) ----------------
__global__ void k_scatter(const float* __restrict__ h, const float* __restrict__ ef,
                          const int* __restrict__ src, const int* __restrict__ dst,
                          const float* __restrict__ g_e, const float* __restrict__ al_e,
                          const float* __restrict__ sum_dst, float* __restrict__ hcat, int E) {
    int e = blockIdx.x * (blockDim.x >> 5) + (threadIdx.x >> 5);
    int l = threadIdx.x & 31;
    if (e >= E) return;
    int s = src[e], d = dst[e];
    float w = al_e[e] / sum_dst[d];
    float gamma = g_e[e];
    const float* hs = h + (size_t)s * 64;
    float* row = hcat + (size_t)d * 160 + 64;
    atomicAdd(row + l,        w * hs[l]);
    atomicAdd(row + 32 + l,   w * hs[32 + l]);
    atomicAdd(row + 64 + l,   w * gamma * ef[(size_t)e * 32 + l]);
}

// ---------------- K6: WMMA f32 node update: out = relu(h_cat @ w_lin) ----------------
// Block = 128 threads (4 waves), one 16-row node block per block; wave w -> output cols [16w,16w+16).
// A tile (16x160) staged into LDS with padded stride 164 (bank-conflict-free WMMA reads).
// Staging path: TDM TENSOR_LOAD_TO_LDS with iterate mode (16 iterations of a 1x160 row,
// lds_addr_increment=164 elements reproduces the pad), tracked by TENSORcnt.
__launch_bounds__(128)
__global__ void k_node_update(const float* __restrict__ hcat, const float* __restrict__ wl,
                              float* __restrict__ out) {
    __shared__ float As[16 * 164];
    const int t = threadIdx.x;
    const int base = blockIdx.x * 16;

#if USE_TDM
    if (t < 32) {                                        // one TDM issue from wave 0 (EXEC ignored by TDM)
        unsigned lds_off = (unsigned)(size_t)(void*)As;  // generic LDS ptr: low 32 bits = LDS byte offset
        unsigned long long ga = (unsigned long long)(const void*)(hcat + (size_t)base * 160);
        // ---- D# group 0: count=1 | lds_addr | global_addr[56:0] | type=2 ----
        u32x4 g0;
        g0[0] = 1u;                                              // count=1, user descriptor
        g0[1] = lds_off;                                         // lds_addr (bytes)
        g0[2] = (unsigned)(ga & 0xFFFFFFFFull);                  // global_addr[31:0]
        g0[3] = (unsigned)((ga >> 32) & 0x01FFFFFFull)           // global_addr[56:32]
              | 0x80000000u;                                     // type=2 ("image") in bits[127:126]
        // ---- D# group 1: data_size=4B, iterate_enable; tensor_dim0=160, tile 160x1, stride 160 ----
        i32x8 g1;
        g1[0] = (2 << 16) | (1 << 19);        // data_size=2(4B), iterate_enable=1, no pad/multicast
        g1[1] = (int)(160u << 16);            // tensor_dim0[15:0] in bits[63:48]
        g1[2] = (int)(16u << 16);             // tensor_dim0 hi=0; tensor_dim1[15:0]=16 in bits[95:80]
        g1[3] = (int)(160u << 16);            // tensor_dim1 hi=0; tile_dim0=160 in bits[127:112]
        g1[4] = 1;                            // tile_dim1=1 (one row per iteration), tile_dim2=0
        g1[5] = 160;                          // tensor_dim0_stride[31:0] = 160 elements
        g1[6] = 0;                            // stride hi + tensor_dim1_stride lo
        g1[7] = 0;
        // ---- D# group 2 (iterate mode): lds += 164 elts, global += 160 elts, 16 iterations ----
        i32x4 g2;
        g2[0] = 1;                            // tensor_dim2 (unused, tile_dim2=0)
        g2[1] = 164;                          // lds_addr_increment (elements) -> padded row stride
        g2[2] = 160;                          // global_addr_increment[31:0] (elements)
        g2[3] = (int)(15u << 16);             // increment hi=0; iterate_count=15 -> 16 rows
        i32x4 g3 = {0, 0, 0, 0};
        i32x8 gx = {0, 0, 0, 0, 0, 0, 0, 0};  // extra descriptor group (6-arg toolchain), unused -> 0
        __builtin_amdgcn_tensor_load_to_lds(g0, g1, g2, g3, gx, 0);
        __builtin_amdgcn_s_wait_tensorcnt(0); // TENSORcnt is per-wave; only wave 0 issued
    }
    __syncthreads();
#else
    for (int i = t; i < 16 * 160; i += 128) {
        int r = i / 160, c = i - r * 160;
        As[r * 164 + c] = hcat[(size_t)(base + r) * 160 + c];
    }
    __syncthreads();
#endif

    const int wave = t >> 5;                 // 0..3 -> N tile
    const int l = t & 31;
    const int l15 = l & 15;
    const int koff = (l < 16) ? 0 : 2;       // A frag: VGPR0=K0/K2, VGPR1=K1/K3 (ISA 16x4 layout)
    const float* Arow = As + l15 * 164;
    const int ncol = wave * 16 + l15;

    v8f c = {};
    #pragma unroll
    for (int k0 = 0; k0 < 160; k0 += 4) {
        v2f a, b;
        a.x = Arow[k0 + koff];
        a.y = Arow[k0 + koff + 1];
        b.x = wl[(size_t)(k0 + koff) * 64 + ncol];       // B 4x16: row K striped across lanes
        b.y = wl[(size_t)(k0 + koff + 1) * 64 + ncol];
        c = __builtin_amdgcn_wmma_f32_16x16x4_f32(false, a, false, b, (short)0, c, false, false);
    }

    const int rowoff = (l < 16) ? 0 : 8;     // C/D layout: VGPR i = rows i and i+8
    #pragma unroll
    for (int i = 0; i < 8; ++i) {
        float v = c[i];
        out[(size_t)(base + i + rowoff) * 64 + ncol] = v > 0.f ? v : 0.f;
    }
}

extern "C" void kernel_launch(void* const* d_in, const int* in_sizes, int n_in,
                              void* d_out, int out_size, void* d_ws, size_t ws_size,
                              hipStream_t stream) {
    const float* h  = (const float*)d_in[0];   // [N,64]
    const float* ef = (const float*)d_in[1];   // [E,32]
    const int* src  = (const int*)d_in[2];     // [E]
    const int* dst  = (const int*)d_in[3];     // [E]
    const float* wg = (const float*)d_in[4];   // [160]
    const float* wa = (const float*)d_in[5];   // [96]
    const float* wl = (const float*)d_in[6];   // [160,64]
    float* out = (float*)d_out;                // [N,64]

    const int N = N_NODES, E = N_EDGES;

    char* p = (char*)d_ws;
    float*    a_e     = (float*)p;    p += (size_t)E * 4;
    float*    g_e     = (float*)p;    p += (size_t)E * 4;   // exp -> gamma (in place)
    float*    s_e     = (float*)p;    p += (size_t)E * 4;
    float*    al_e    = (float*)p;    p += (size_t)E * 4;
    unsigned* m_src   = (unsigned*)p; p += (size_t)N * 4;
    float*    sum_src = (float*)p;    p += (size_t)N * 4;
    unsigned* m_dst   = (unsigned*)p; p += (size_t)N * 4;
    float*    sum_dst = (float*)p;    p += (size_t)N * 4;
    float*    hcat    = (float*)p;    // [N,160]

    const int wavesPerBlk = 256 / 32;                // 8 edges per 256-thread block
    const int edgeWaveGrid = (E + wavesPerBlk - 1) / wavesPerBlk;   // 100000
    const int edgeThrGrid  = (E + 255) / 256;                       // 3125
    const int nodeGrid     = (N + 255) / 256;

    k_init<<<nodeGrid, 256, 0, stream>>>(h, m_src, sum_src, m_dst, sum_dst, hcat, N);
    k_src_score<<<edgeWaveGrid, 256, 0, stream>>>(h, ef, src, dst, wg, a_e, m_src, E);
    k_seg_exp<<<edgeThrGrid, 256, 0, stream>>>(a_e, src, m_src, g_e, sum_src, E);
    k_msg_score<<<edgeWaveGrid, 256, 0, stream>>>(h, ef, src, dst, wa, sum_src, g_e, s_e, m_dst, E);
    k_seg_exp<<<edgeThrGrid, 256, 0, stream>>>(s_e, dst, m_dst, al_e, sum_dst, E);
    k_scatter<<<edgeWaveGrid, 256, 0, stream>>>(h, ef, src, dst, g_e, al_e, sum_dst, hcat, E);
    k_node_update<<<N / 16, 128, 0, stream>>>(hcat, wl, out);   // 3125 blocks, 40 WMMAs/tile
}